// Dense_Adapter_gai_62886911148137
// MI455X (gfx1250) — compile-verified
//
#include <hip/hip_runtime.h>

typedef __attribute__((ext_vector_type(2))) float v2f;
typedef __attribute__((ext_vector_type(8))) float v8f;
typedef __attribute__((ext_vector_type(4))) int   v4i;

#define DIMX 768
#define CAX  192
#define LSEQ 197
#define BTX  128
#define TT   8
#define KC   32        // K-chunk staged in LDS (32 x 192 floats = 24 KB per buffer)

#if __has_builtin(__builtin_amdgcn_global_load_async_to_lds_b128) && \
    __has_builtin(__builtin_amdgcn_s_wait_asynccnt)
#define USE_ASYNC_LDS 1
typedef __attribute__((address_space(1))) v4i* gv4i_p;   // global (device) ptr
typedef __attribute__((address_space(3))) v4i* lv4i_p;   // LDS ptr
#else
#define USE_ASYNC_LDS 0
#endif

__device__ __forceinline__ float gelu_exact(float v) {
    return 0.5f * v * (1.0f + erff(v * 0.70710678118654752f));
}

// Copy one 16-byte element (float4) from global to LDS, async when available.
__device__ __forceinline__ void copy16_to_lds(const float* gsrc, float* ldst) {
#if USE_ASYNC_LDS
    __builtin_amdgcn_global_load_async_to_lds_b128(
        (gv4i_p)(void*)gsrc, (lv4i_p)(void*)ldst, 0, 0);
#else
    *(float4*)ldst = *(const float4*)gsrc;
#endif
}

__device__ __forceinline__ void wait_lds_fill() {
#if USE_ASYNC_LDS
    __builtin_amdgcn_s_wait_asynccnt(0);
#endif
}

// ---------------------------------------------------------------------------
// Kernel 1: h = x @ w1 + b1 fused with temporal conv / diff-mix / GELU.
// Only rows l >= 1 are needed (reference discards l==0 of h).
// Block = 256 thr (8 waves); W1 K-chunks (32 x 192) double-buffered in LDS:
// async fill of chunk i+1 overlaps WMMA compute of chunk i.
// One wave computes a 16-row x 192-col tile with V_WMMA_F32_16X16X4_F32.
// C/D layout: lane in [0,32), col = 16*ct + (lane&15),
//             row = row0 + 8*(lane>>4) + v  ->  t = row % 8 = v (lane-local!)
// ---------------------------------------------------------------------------
__global__ __launch_bounds__(256) void gemm1_mix_gelu(
    const float* __restrict__ X,      // (25216, 768), rows r = l*128 + bt
    const float* __restrict__ W1,     // (768, 192)
    const float* __restrict__ B1,     // (192)
    const float* __restrict__ CW,     // (192, 3)
    const float* __restrict__ CB,     // (192)
    float* __restrict__ G)            // (25216, 192) activation out
{
    __shared__ __align__(16) float sB[2][KC * CAX];   // 2 x 24 KB

    const int lane  = threadIdx.x & 31;
    const int wave  = threadIdx.x >> 5;
    const int gwave = blockIdx.x * 8 + wave;      // 0 .. 1567
    const int row0  = 128 + gwave * 16;           // rows for l = 1..196
    const int m     = lane & 15;
    const int kh    = lane >> 4;

    v8f acc[12];
    #pragma unroll
    for (int i = 0; i < 12; ++i)
        #pragma unroll
        for (int j = 0; j < 8; ++j) acc[i][j] = 0.0f;

    const float* arow = X + (size_t)(row0 + m) * DIMX;

    // Prologue: kick off async fill of chunk 0 into buffer 0.
    {
        const float* src = W1;                    // chunk kc=0 is contiguous 24 KB
        #pragma unroll
        for (int i = 0; i < 6; ++i) {
            const int e = threadIdx.x + i * 256;  // float4 index < 1536
            copy16_to_lds(src + e * 4, &sB[0][e * 4]);
        }
    }

    int p = 0;
    for (int kc = 0; kc < DIMX; kc += KC) {
        wait_lds_fill();                          // own fills of buffer p done
        __syncthreads();                          // all fills visible; prev buf free
        if (kc + KC < DIMX) {                     // overlap: fill next chunk
            const float* src = W1 + (size_t)(kc + KC) * CAX;
            float* dst = &sB[p ^ 1][0];
            #pragma unroll
            for (int i = 0; i < 6; ++i) {
                const int e = threadIdx.x + i * 256;
                copy16_to_lds(src + e * 4, dst + e * 4);
            }
        }
        __builtin_prefetch(arow + kc + KC, 0, 1); // stream next A chunk

        const float* buf = &sB[p][0];
        #pragma unroll
        for (int k8 = 0; k8 < KC; k8 += 4) {
            v2f a;
            a.x = arow[kc + k8 + 2 * kh];
            a.y = arow[kc + k8 + 2 * kh + 1];
            const float* b0  = buf + (k8 + 2 * kh) * CAX;
            const float* b1p = b0 + CAX;
            #pragma unroll
            for (int ct = 0; ct < 12; ++ct) {
                const int col = ct * 16 + m;
                v2f b;
                b.x = b0[col];
                b.y = b1p[col];
                acc[ct] = __builtin_amdgcn_wmma_f32_16x16x4_f32(
                    false, a, false, b, (short)0, acc[ct], false, false);
            }
        }
        p ^= 1;
    }

    // Epilogue: bias + temporal mixing + GELU, entirely lane-local.
    const int rbase  = row0 + (kh << 3);          // first of this lane-half's 8 rows
    const bool is_l1 = (row0 < 256);              // l == 1 tiles also emit CLS rows

    #pragma unroll
    for (int ct = 0; ct < 12; ++ct) {
        const int col = ct * 16 + m;
        const float bv  = B1[col];
        const float cw0 = CW[col * 3 + 0];
        const float cw1 = CW[col * 3 + 1];
        const float cw2 = CW[col * 3 + 2];
        const float cb  = CB[col];

        float xt[TT];
        #pragma unroll
        for (int t = 0; t < TT; ++t) xt[t] = acc[ct][t] + bv;

        // S = sum_t [ t * x_after[t] - (7-t) * xt[t] ],
        // x_after[t] = cw0*xt[t-1] + cw1*xt[t] + cw2*xt[t+1] + cb (zero-padded)
        float S = 0.0f;
        #pragma unroll
        for (int t = 0; t < TT; ++t) {
            const float prev = (t > 0)      ? xt[t - 1] : 0.0f;
            const float next = (t < TT - 1) ? xt[t + 1] : 0.0f;
            const float xa = cw0 * prev + cw1 * xt[t] + cw2 * next + cb;
            S += (float)t * xa - (float)(TT - 1 - t) * xt[t];
        }
        const float res = S * (1.0f / 28.0f);

        #pragma unroll
        for (int t = 0; t < TT; ++t) {
            const size_t r = (size_t)(rbase + t);
            G[r * CAX + col] = gelu_exact(xt[t] + res);
            if (is_l1) {
                // CLS output row (l==0) = gelu(h[l==1]) before mixing
                G[(r - 128) * CAX + col] = gelu_exact(xt[t]);
            }
        }
    }
}

// ---------------------------------------------------------------------------
// Kernel 2: out = x + g @ w2 + b2.  M=25216, K=192, N=768.
// Grid (197, 4): blockIdx.y = 192-col group; 8 waves = 8 row-tiles per block.
// W2 K-chunks (32 rows x this block's 192 cols) double-buffered in LDS.
// ---------------------------------------------------------------------------
__global__ __launch_bounds__(256) void gemm2_resid(
    const float* __restrict__ G,      // (25216, 192)
    const float* __restrict__ W2,     // (192, 768)
    const float* __restrict__ B2,     // (768)
    const float* __restrict__ X,      // (25216, 768) residual
    float* __restrict__ O)            // (25216, 768)
{
    __shared__ __align__(16) float sB[2][KC * CAX];   // 2 x 24 KB

    const int lane  = threadIdx.x & 31;
    const int wave  = threadIdx.x >> 5;
    const int gwave = blockIdx.x * 8 + wave;      // 0 .. 1575 (row tile)
    const int row0  = gwave * 16;
    const int col0  = blockIdx.y * 192;
    const int m     = lane & 15;
    const int kh    = lane >> 4;

    v8f acc[12];
    #pragma unroll
    for (int i = 0; i < 12; ++i)
        #pragma unroll
        for (int j = 0; j < 8; ++j) acc[i][j] = 0.0f;

    const float* arow = G + (size_t)(row0 + m) * CAX;

    // Prologue: async fill of chunk 0 (rows 0..31 of W2, cols col0..col0+192).
    #pragma unroll
    for (int i = 0; i < 6; ++i) {
        const int e   = threadIdx.x + i * 256;    // float4 index < 1536
        const int row = e / 48;                   // 48 float4 per row
        const int c4  = e % 48;
        copy16_to_lds(W2 + (size_t)row * DIMX + col0 + c4 * 4,
                      &sB[0][row * CAX + c4 * 4]);
    }

    int p = 0;
    for (int kc = 0; kc < CAX; kc += KC) {
        wait_lds_fill();
        __syncthreads();
        if (kc + KC < CAX) {
            float* dst = &sB[p ^ 1][0];
            #pragma unroll
            for (int i = 0; i < 6; ++i) {
                const int e   = threadIdx.x + i * 256;
                const int row = e / 48;
                const int c4  = e % 48;
                copy16_to_lds(W2 + (size_t)(kc + KC + row) * DIMX + col0 + c4 * 4,
                              dst + row * CAX + c4 * 4);
            }
        }
        __builtin_prefetch(arow + kc + KC, 0, 1);

        const float* buf = &sB[p][0];
        #pragma unroll
        for (int k8 = 0; k8 < KC; k8 += 4) {
            v2f a;
            a.x = arow[kc + k8 + 2 * kh];
            a.y = arow[kc + k8 + 2 * kh + 1];
            const float* b0  = buf + (k8 + 2 * kh) * CAX;
            const float* b1p = b0 + CAX;
            #pragma unroll
            for (int ct = 0; ct < 12; ++ct) {
                const int col = ct * 16 + m;
                v2f b;
                b.x = b0[col];
                b.y = b1p[col];
                acc[ct] = __builtin_amdgcn_wmma_f32_16x16x4_f32(
                    false, a, false, b, (short)0, acc[ct], false, false);
            }
        }
        p ^= 1;
    }

    const int rbase = row0 + (kh << 3);
    #pragma unroll
    for (int ct = 0; ct < 12; ++ct) {
        const int col = col0 + ct * 16 + m;
        const float bv = B2[col];
        #pragma unroll
        for (int v = 0; v < 8; ++v) {
            const size_t idx = (size_t)(rbase + v) * DIMX + col;
            O[idx] = X[idx] + acc[ct][v] + bv;
        }
    }
}

extern "C" void kernel_launch(void* const* d_in, const int* in_sizes, int n_in,
                              void* d_out, int out_size, void* d_ws, size_t ws_size,
                              hipStream_t stream) {
    const float* x      = (const float*)d_in[0];
    const float* w1     = (const float*)d_in[1];
    const float* b1     = (const float*)d_in[2];
    const float* conv_w = (const float*)d_in[3];
    const float* conv_b = (const float*)d_in[4];
    const float* w2     = (const float*)d_in[5];
    const float* b2     = (const float*)d_in[6];
    float* out = (float*)d_out;
    float* g   = (float*)d_ws;   // 25216 * 192 floats = 19.4 MB scratch

    // GEMM1 fused: 1568 row-tiles (l=1..196), 8 waves/block -> 196 blocks
    gemm1_mix_gelu<<<196, 256, 0, stream>>>(x, w1, b1, conv_w, conv_b, g);
    // GEMM2 fused: 1576 row-tiles x 4 col-groups, 8 row-tiles/block
    gemm2_resid<<<dim3(197, 4), 256, 0, stream>>>(g, w2, b2, x, out);
}